// SupContLoss_general_23828478558276
// MI455X (gfx1250) — compile-verified
//
#include <hip/hip_runtime.h>
#include <math.h>

typedef float v2f __attribute__((ext_vector_type(2)));
typedef float v8f __attribute__((ext_vector_type(8)));

#define TEMP      0.07f
#define INV_TEMP  (1.0f / 0.07f)
#define NCON      20
#define LDIM      512
#define NROWS     32768
#define NSEL      16384

// workspace layout (float offsets)
#define WS_E      0                      // e_sz[512] then e_nsz[512] (contiguous)
#define WS_KSZ    1024                   // keep_sz[20]
#define WS_KNSZ   1056                   // keep_nsz[20]
#define WS_DSZ    2048                   // dot(h, e_sz) per row [32768]
#define WS_DNSZ   (2048 + NROWS)         // dot(h, e_nsz) per row [32768]
#define WS_SS     (2048 + 2 * NROWS)     // sum of squares per row [32768]
#define WS_PART   (2048 + 3 * NROWS)     // per-block partials [128]

// ---------------------------------------------------------------------------
// Kernel 0: e_bar vectors (mean of 5 selected concept embeddings) + keep masks
// ---------------------------------------------------------------------------
__global__ __launch_bounds__(512) void prep_kernel(const float* __restrict__ all_emb,
                                                   const int* __restrict__ Psz,
                                                   const int* __restrict__ Pnsz,
                                                   float* ws) {
    const int t = threadIdx.x;  // 0..511
    float esz = 0.f, ensz = 0.f;
    for (int p = 0; p < 5; ++p) {
        esz  += all_emb[Psz[p]  * LDIM + t];
        ensz += all_emb[Pnsz[p] * LDIM + t];
    }
    ws[WS_E + t]        = esz  * 0.2f;
    ws[WS_E + LDIM + t] = ensz * 0.2f;
    if (t < NCON) {
        float ks = 1.f, kn = 1.f;
        for (int p = 0; p < 5; ++p) {
            if (Psz[p]  == t) ks = 0.f;
            if (Pnsz[p] == t) kn = 0.f;
        }
        ws[WS_KSZ  + t] = ks;
        ws[WS_KNSZ + t] = kn;
    }
}

// ---------------------------------------------------------------------------
// Kernel 1: single streaming pass over hg. One wave owns a 16-row tile and
// accumulates D = A(16x4) x B(4x16) over K=512 with V_WMMA_F32_16X16X4_F32.
// B col 0 = e_sz, col 1 = e_nsz, cols 2..15 = 0. Row sum-of-squares is
// accumulated from the same A loads.
// ---------------------------------------------------------------------------
__global__ __launch_bounds__(256) void dot_norm_kernel(const float* __restrict__ hg,
                                                       float* ws) {
    __shared__ float s_e[2 * LDIM];  // e_sz | e_nsz
    for (int i = threadIdx.x; i < 2 * LDIM; i += 256)
        s_e[i] = ws[WS_E + i];
    __syncthreads();

    const int wave = threadIdx.x >> 5;
    const int lane = threadIdx.x & 31;
    const int tile = blockIdx.x * 8 + wave;   // 2048 tiles total
    const int rb   = tile * 16;               // first row of tile
    const int M    = lane & 15;               // row within tile / B column
    const int kb   = (lane >> 4) << 1;        // 0 (lanes 0-15) or 2 (lanes 16-31)

    const float* aptr  = hg + (size_t)(rb + M) * LDIM + kb;
    const float* bbase = (M == 0) ? &s_e[0] : &s_e[LDIM];  // lane's B column source
    const bool   buse  = (M < 2);

    v8f c = {0.f, 0.f, 0.f, 0.f, 0.f, 0.f, 0.f, 0.f};
    float ss = 0.f;
    for (int k0 = 0; k0 < LDIM; k0 += 4) {
        v2f a  = *(const v2f*)(aptr + k0);          // A[M][k0+kb .. +1]
        v2f bv = *(const v2f*)(bbase + k0 + kb);    // e[k0+kb .. +1]
        v2f b;
        b.x = buse ? bv.x : 0.f;   // cndmask: no EXEC divergence
        b.y = buse ? bv.y : 0.f;
        ss += a.x * a.x + a.y * a.y;
        // D = A x B + C  (fp32 WMMA, 8-arg form)
        c = __builtin_amdgcn_wmma_f32_16x16x4_f32(false, a, false, b,
                                                  (short)0, c, false, false);
    }

    // complete per-row sum of squares: lane l covers cols {kb,kb+1} mod 4
    float ss2 = ss + __shfl_xor(ss, 16);
    if (lane < 16) ws[WS_SS + rb + M] = ss2;

    // D layout: lane = column N (0-15 rows M=0..7 in c[0..7]; 16-31 rows 8..15)
    float* dsz  = ws + WS_DSZ;
    float* dnsz = ws + WS_DNSZ;
    if (lane == 0)  { for (int r = 0; r < 8; ++r) dsz [rb + r]     = c[r]; }
    if (lane == 16) { for (int r = 0; r < 8; ++r) dsz [rb + 8 + r] = c[r]; }
    if (lane == 1)  { for (int r = 0; r < 8; ++r) dnsz[rb + r]     = c[r]; }
    if (lane == 17) { for (int r = 0; r < 8; ++r) dnsz[rb + 8 + r] = c[r]; }
}

// ---------------------------------------------------------------------------
// Kernel 2: per-selected-row loss term + deterministic block reduction.
// 128 blocks x 256 threads: blocks 0-63 -> sz branch, 64-127 -> nsz branch.
// ---------------------------------------------------------------------------
__global__ __launch_bounds__(256) void loss_kernel(const float* __restrict__ corr,
                                                   const int* __restrict__ sz_idx,
                                                   const int* __restrict__ nsz_idx,
                                                   float* ws) {
    const int g      = blockIdx.x * 256 + threadIdx.x;
    const int branch = g >> 14;            // uniform per block
    const int j      = g & (NSEL - 1);

    const int*   idx  = branch ? nsz_idx : sz_idx;
    const float* dot  = ws + (branch ? WS_DNSZ : WS_DSZ);
    const float* keep = ws + (branch ? WS_KNSZ : WS_KSZ);

    const int i = idx[j];
    const float nrm = fmaxf(sqrtf(ws[WS_SS + i]), 1e-12f);
    const float d   = dot[i] / nrm;

    float den = 0.f;
    const float* cr = corr + (size_t)i * NCON;
    for (int cidx = 0; cidx < NCON; ++cidx)
        den += keep[cidx] * expf(cr[cidx] * INV_TEMP);

    float val = -d * INV_TEMP + logf(den);

    __shared__ float red[256];
    red[threadIdx.x] = val;
    __syncthreads();
    for (int s = 128; s > 0; s >>= 1) {
        if (threadIdx.x < s) red[threadIdx.x] += red[threadIdx.x + s];
        __syncthreads();
    }
    if (threadIdx.x == 0) ws[WS_PART + blockIdx.x] = red[0];
}

// ---------------------------------------------------------------------------
// Kernel 3: final deterministic reduction of 128 partials -> scalar.
// total = (sum_sz + sum_nsz) / 16384
// ---------------------------------------------------------------------------
__global__ __launch_bounds__(128) void final_kernel(const float* __restrict__ ws,
                                                    float* __restrict__ out) {
    __shared__ float red[128];
    red[threadIdx.x] = ws[WS_PART + threadIdx.x];
    __syncthreads();
    for (int s = 64; s > 0; s >>= 1) {
        if (threadIdx.x < s) red[threadIdx.x] += red[threadIdx.x + s];
        __syncthreads();
    }
    if (threadIdx.x == 0) out[0] = red[0] * (1.0f / (float)NSEL);
}

extern "C" void kernel_launch(void* const* d_in, const int* in_sizes, int n_in,
                              void* d_out, int out_size, void* d_ws, size_t ws_size,
                              hipStream_t stream) {
    const float* hg       = (const float*)d_in[0];  // [32768, 512]
    const float* corr     = (const float*)d_in[1];  // [32768, 20]
    const float* all_emb  = (const float*)d_in[2];  // [20, 512]
    const int*   sz_idx   = (const int*)d_in[3];    // [16384]
    const int*   nsz_idx  = (const int*)d_in[4];    // [16384]
    const int*   Psz_idx  = (const int*)d_in[5];    // [5]
    const int*   Pnsz_idx = (const int*)d_in[6];    // [5]
    float* out = (float*)d_out;
    float* ws  = (float*)d_ws;

    prep_kernel<<<1, 512, 0, stream>>>(all_emb, Psz_idx, Pnsz_idx, ws);
    dot_norm_kernel<<<256, 256, 0, stream>>>(hg, ws);           // 2048 wave-tiles
    loss_kernel<<<128, 256, 0, stream>>>(corr, sz_idx, nsz_idx, ws);
    final_kernel<<<1, 128, 0, stream>>>(ws, out);
}